// FSA_RNN_10582799417526
// MI455X (gfx1250) — compile-verified
//
#include <hip/hip_runtime.h>

// ---------------------------------------------------------------------------
// FSA-RNN on MI455X (gfx1250).
//
// h_L = T_L ... T_1 h0 with T_t = relu(W[x_t]) (16x16).  Matrix product is
// associative, so:
//   Phase 1: 512 batches x 16 chunks of 128 steps -> 8192 independent waves,
//            each computing a chunk product via chained V_WMMA_F32_16X16X4_F32
//            (4 K=4 WMMAs per 16x16x16 f32 matmul).  relu(W) lives in LDS
//            (288 KB, padded rows) -- only possible on CDNA5's 320 KB/WGP LDS.
//   Phase 2: per batch, fold the 16 chunk matrices into relu(start), dot with
//            final_states.  ~2 MFLOP, scalar.
//
// Serial-chain hygiene: token broadcast uses v_readlane (SGPR lane select,
// uniform result) so token math runs on SALU, keeping the DS pipe free for
// the A/B operand loads that feed the WMMA chain.
// ---------------------------------------------------------------------------

typedef float v2f __attribute__((ext_vector_type(2)));
typedef float v8f __attribute__((ext_vector_type(8)));

#define SDIM       16
#define BATCH      512
#define SEQLEN     2048
#define VOCAB      256
#define CHUNKS     16
#define CHUNK_LEN  (SEQLEN / CHUNKS)     // 128
#define WAVES_PB   8                     // waves per block (256 threads, wave32)
#define ROWPAD     18                    // padded row stride (floats): 18*l mod 64 distinct
#define WSTRIDE    (SDIM * ROWPAD)       // 288 floats per padded 16x16 matrix

// ---------------------------------------------------------------------------
// Phase 1: chunk products.  One wave per (batch, chunk).
//
// Register layouts (wave32):
//   C/D (16x16 f32, 8 VGPRs): c[v] = P[v + 8*(lane>=16)][lane&15]
//   A slice k (16x4):  vgpr j = T[lane&15][4k + j + 2*(lane>>4)]
//   B slice k (4x16):  vgpr j = P[4k + j + 2*(lane>>4)][lane&15]
// T is read from LDS row-major (A layout = contiguous float2 per lane).
// P is spilled column-major so its B-layout read is a contiguous float2.
// ---------------------------------------------------------------------------
__global__ __launch_bounds__(256)
void fsa_chunk_products(const int* __restrict__ x,
                        const float* __restrict__ W,
                        float* __restrict__ ws) {
    extern __shared__ float smem[];
    float* ldsW = smem;                          // VOCAB * WSTRIDE floats (288 KB)
    float* ldsP = smem + VOCAB * WSTRIDE;        // WAVES_PB * WSTRIDE floats

    const int tid = threadIdx.x;

    // ---- cooperative fill: relu(W) -> padded LDS (b128 global loads) ----
    for (int f4 = tid; f4 < (VOCAB * SDIM * SDIM) / 4; f4 += 256) {
        const float4 w4 = ((const float4*)W)[f4];
        const int f  = f4 * 4;
        const int v  = f >> 8;
        const int r  = (f >> 4) & 15;
        const int c0 = f & 15;
        float* dst = &ldsW[v * WSTRIDE + r * ROWPAD + c0];
        dst[0] = fmaxf(w4.x, 0.f);
        dst[1] = fmaxf(w4.y, 0.f);
        dst[2] = fmaxf(w4.z, 0.f);
        dst[3] = fmaxf(w4.w, 0.f);
    }
    __syncthreads();

    const int wave = tid >> 5;
    const int lane = tid & 31;
    const int m    = lane & 15;       // row (A) or column (B, C/D)
    const int hh   = lane >> 4;       // lane half

    const int task  = blockIdx.x * WAVES_PB + wave;   // 0..8191
    const int b     = task >> 4;
    const int chunk = task & 15;
    const int* xb   = x + b * SEQLEN + chunk * CHUNK_LEN;

    __builtin_prefetch(xb, 0, 0);     // gfx1250 global_prefetch_b8 on token stream

    // preload this chunk's 128 tokens, 4 per lane, coalesced
    int tk[CHUNK_LEN / 32];
#pragma unroll
    for (int i = 0; i < CHUNK_LEN / 32; ++i) tk[i] = xb[i * 32 + lane];

    float* Pbuf = &ldsP[wave * WSTRIDE];

    // loop-invariant per-lane offsets
    const int aOff = m * ROWPAD + 2 * hh;          // A-slice base within a W tile
    const float* Pcol      = &Pbuf[m * ROWPAD + 2 * hh];   // B-slice read base
    float*       Pspill    = &Pbuf[m * ROWPAD + 8 * hh];   // column-major spill base

    // P = relu(W[x[t0]]) straight into C/D register layout
    {
        const int tok0 = __builtin_amdgcn_readlane(tk[0], 0);
        const float* T0 = &ldsW[tok0 * WSTRIDE];
        v8f a0;
#pragma unroll
        for (int v = 0; v < 8; ++v)
            a0[v] = T0[(v + 8 * hh) * ROWPAD + m];

        // sequential chain: P <- T_t * P  (4 chained K=4 WMMAs per step)
        v8f acc = a0;
#pragma unroll
        for (int blk = 0; blk < CHUNK_LEN / 32; ++blk) {
            const int tkblk = tk[blk];       // fixed register per (unrolled) block
#pragma unroll 4
            for (int ss = (blk == 0 ? 1 : 0); ss < 32; ++ss) {
                // spill P column-major (padded, 8B-aligned) for B-layout reread;
                // same-wave LDS ops are in-order
#pragma unroll
                for (int u = 0; u < 8; u += 2)
                    *(float2*)&Pspill[u] = make_float2(acc[u], acc[u + 1]);

                // uniform token -> SALU address math, DS pipe stays free
                const int tok = __builtin_amdgcn_readlane(tkblk, ss);
                const float* Trow = &ldsW[tok * WSTRIDE] + aOff;

                v8f d = {};                      // fresh accumulator (inline 0 C)
#pragma unroll
                for (int k = 0; k < 4; ++k) {
                    const float2 a2 = *(const float2*)&Trow[4 * k];
                    const float2 b2 = *(const float2*)&Pcol[4 * k];
                    v2f av; av[0] = a2.x; av[1] = a2.y;
                    v2f bv; bv[0] = b2.x; bv[1] = b2.y;
                    // D = A_k * B_k + D  (neg_a, A, neg_b, B, c_mod, C, reuse_a/b)
                    d = __builtin_amdgcn_wmma_f32_16x16x4_f32(
                            false, av, false, bv, (short)0, d, false, false);
                }
                acc = d;
            }
        }

        // write chunk product row-major to workspace
        float* out = ws + (size_t)task * (SDIM * SDIM);
#pragma unroll
        for (int v = 0; v < 8; ++v)
            out[(v + 8 * hh) * SDIM + m] = acc[v];
    }
}

// ---------------------------------------------------------------------------
// Phase 2: fold 16 chunk matrices into relu(start), dot with final.
// 16 batches per block, 16 threads per batch (thread i owns state i).
// ---------------------------------------------------------------------------
__global__ __launch_bounds__(256)
void fsa_combine(const float* __restrict__ ws,
                 const float* __restrict__ start,
                 const float* __restrict__ fin,
                 float* __restrict__ out) {
    __shared__ float hbuf[16][SDIM];
    const int tid = threadIdx.x;
    const int bl  = tid >> 4;              // local batch
    const int i   = tid & 15;              // state index
    const int b   = blockIdx.x * 16 + bl;

    hbuf[bl][i] = fmaxf(start[i], 0.f);
    __syncthreads();

    for (int c = 0; c < CHUNKS; ++c) {
        const float* M = ws + ((size_t)(b * CHUNKS + c)) * (SDIM * SDIM) + i * SDIM;
        float s = 0.f;
#pragma unroll
        for (int j = 0; j < SDIM; ++j) s += M[j] * hbuf[bl][j];
        __syncthreads();
        hbuf[bl][i] = s;
        __syncthreads();
    }

    const float partial = hbuf[bl][i] * fin[i];
    __syncthreads();
    hbuf[bl][i] = partial;
    __syncthreads();
    if (i == 0) {
        float s = 0.f;
#pragma unroll
        for (int j = 0; j < SDIM; ++j) s += hbuf[bl][j];
        out[b] = s;
    }
}

// ---------------------------------------------------------------------------
extern "C" void kernel_launch(void* const* d_in, const int* in_sizes, int n_in,
                              void* d_out, int out_size, void* d_ws, size_t ws_size,
                              hipStream_t stream) {
    const int*   x     = (const int*)d_in[0];      // [512, 2048] int32
    const float* W     = (const float*)d_in[1];    // [256, 256]  f32
    const float* start = (const float*)d_in[2];    // [16]        f32
    const float* fin   = (const float*)d_in[3];    // [16]        f32
    float*       out   = (float*)d_out;            // [512]       f32
    float*       ws    = (float*)d_ws;             // needs 8192*256*4 = 8 MiB

    const size_t lds_bytes =
        (size_t)(VOCAB * WSTRIDE + WAVES_PB * WSTRIDE) * sizeof(float); // ~297 KB

    dim3 grid1(BATCH * CHUNKS / WAVES_PB);   // 1024 blocks x 8 waves = 8192 tasks
    fsa_chunk_products<<<grid1, 256, lds_bytes, stream>>>(x, W, ws);

    fsa_combine<<<BATCH / 16, 256, 0, stream>>>(ws, start, fin, out);
}